// MMDLoss_5935644803544
// MI455X (gfx1250) — compile-verified
//
#include <hip/hip_runtime.h>

typedef __attribute__((ext_vector_type(2))) float v2f;
typedef __attribute__((ext_vector_type(4))) float v4f;
typedef __attribute__((ext_vector_type(8))) float v8f;

#define NS    4096
#define NTOT  8192
#define KDIM  256
#define TILE  128
#define LD    260          // padded LDS row stride in floats (bank-conflict-free)
#define NSUB  8            // 8 column sub-tiles of 16 per wave

// workspace layout (floats)
#define WS_SQ     0                   // 8192 row squared-norms
#define WS_P1     (WS_SQ + NTOT)      // 4096 per-block partials, pass 1
#define WS_P2     (WS_P1 + 4096)      // 4096 per-block partials, pass 2
#define WS_BWSUM  (WS_P2 + 4096)      // 1: sum(l2)

// ---------------------------------------------------------------- row norms
__global__ __launch_bounds__(256) void rownorm_kernel(const float* __restrict__ src,
                                                      const float* __restrict__ tgt,
                                                      float* __restrict__ ws) {
    int wave = threadIdx.x >> 5, lane = threadIdx.x & 31;
    int row  = blockIdx.x * 8 + wave;
    const float* p = (row < NS) ? (src + (size_t)row * KDIM)
                                : (tgt + (size_t)(row - NS) * KDIM);
    float s = 0.f;
    for (int q = 0; q < KDIM; q += 32) { float x = p[q + lane]; s += x * x; }
    for (int off = 16; off; off >>= 1) s += __shfl_down(s, off, 32);
    if (lane == 0) ws[WS_SQ + row] = s;
}

// ------------------------------------------- shared 128x128 WMMA gram tile
__device__ __forceinline__ void stage_and_gemm(const float* __restrict__ src,
                                               const float* __restrict__ tgt,
                                               float* ldsA, float* ldsB,
                                               int i0, int j0, v8f acc[NSUB]) {
    const float* A = (i0 < NS) ? src + (size_t)i0 * KDIM : tgt + (size_t)(i0 - NS) * KDIM;
    const float* B = (j0 < NS) ? src + (size_t)j0 * KDIM : tgt + (size_t)(j0 - NS) * KDIM;

    // cooperative stage: two 128x256 f32 panels, DMA'd straight into LDS with
    // gfx1250 async loads (ASYNCcnt path, no VGPR round-trip).
    unsigned ldsA_off = (unsigned)(uintptr_t)(void*)ldsA;   // low 32 bits == LDS byte offset
    unsigned ldsB_off = (unsigned)(uintptr_t)(void*)ldsB;
    for (int iter = 0; iter < 32; ++iter) {
        int flat = iter * 1024 + threadIdx.x * 4;   // float index in 128x256 panel
        int r = flat >> 8, c = flat & 255;
        unsigned loff = (unsigned)((r * LD + c) * 4);
        const float* ga = A + (size_t)r * KDIM + c;
        const float* gb = B + (size_t)r * KDIM + c;
        unsigned la = ldsA_off + loff;
        unsigned lb = ldsB_off + loff;
        asm volatile("global_load_async_to_lds_b128 %0, %1, off"
                     :: "v"(la), "v"(ga) : "memory");
        asm volatile("global_load_async_to_lds_b128 %0, %1, off"
                     :: "v"(lb), "v"(gb) : "memory");
    }
    asm volatile("s_wait_asynccnt 0x0" ::: "memory");
    __syncthreads();

    int lane = threadIdx.x & 31, wave = threadIdx.x >> 5;
    int lr = lane & 15;        // row within 16-tile
    int kh = lane >> 4;        // selects K pair {0,1} vs {2,3}

    const float* ap = ldsA + (wave * 16 + lr) * LD + 2 * kh;
    const float* bp = ldsB + lr * LD + 2 * kh;

    v8f zero = {};
#pragma unroll
    for (int t = 0; t < NSUB; ++t) acc[t] = zero;

    for (int k0 = 0; k0 < KDIM; k0 += 4) {
        v2f a = *(const v2f*)(ap + k0);              // A fragment: reused 8x
#pragma unroll
        for (int t = 0; t < NSUB; ++t) {
            v2f b = *(const v2f*)(bp + t * 16 * LD + k0);
            acc[t] = __builtin_amdgcn_wmma_f32_16x16x4_f32(
                false, a, false, b, (short)0, acc[t], false, false);
        }
    }
}

__device__ __forceinline__ void block_reduce_store(float v, float* scratch, float* dst) {
    int lane = threadIdx.x & 31, wave = threadIdx.x >> 5;
    for (int off = 16; off; off >>= 1) v += __shfl_down(v, off, 32);
    __syncthreads();                     // done reading LDS panels before reuse
    if (lane == 0) scratch[wave] = v;
    __syncthreads();
    if (threadIdx.x == 0) {
        float s = 0.f;
#pragma unroll
        for (int w = 0; w < 8; ++w) s += scratch[w];
        *dst = s;
    }
}

// --------------------------------------------------- pass 1: sum of l2
__global__ __launch_bounds__(256) void suml2_kernel(const float* __restrict__ src,
                                                    const float* __restrict__ tgt,
                                                    float* __restrict__ ws) {
    extern __shared__ float lds[];
    float* ldsA = lds;
    float* ldsB = lds + TILE * LD;
    int i0 = blockIdx.y * TILE, j0 = blockIdx.x * TILE;

    v8f acc[NSUB];
    stage_and_gemm(src, tgt, ldsA, ldsB, i0, j0, acc);

    int lane = threadIdx.x & 31, wave = threadIdx.x >> 5;
    int lr = lane & 15, kh = lane >> 4;
    const float* sq = ws + WS_SQ;
    int ibase = i0 + wave * 16 + 8 * kh;
    float sqi[8];
#pragma unroll
    for (int v = 0; v < 8; ++v) sqi[v] = sq[ibase + v];

    float lsum = 0.f;
#pragma unroll
    for (int t = 0; t < NSUB; ++t) {
        float sqj = sq[j0 + t * 16 + lr];
#pragma unroll
        for (int v = 0; v < 8; ++v) {
            float l2 = fmaxf(sqi[v] + sqj - 2.0f * acc[t][v], 0.0f);
            lsum += l2;
        }
    }
    block_reduce_store(lsum, lds, ws + WS_P1 + blockIdx.y * gridDim.x + blockIdx.x);
}

// ------------------------------ pass 2: signed 5-kernel Gaussian sum
__global__ __launch_bounds__(256) void mmd_kernel(const float* __restrict__ src,
                                                  const float* __restrict__ tgt,
                                                  float* __restrict__ ws) {
    extern __shared__ float lds[];
    float* ldsA = lds;
    float* ldsB = lds + TILE * LD;
    int i0 = blockIdx.y * TILE, j0 = blockIdx.x * TILE;

    v8f acc[NSUB];
    stage_and_gemm(src, tgt, ldsA, ldsB, i0, j0, acc);

    int lane = threadIdx.x & 31, wave = threadIdx.x >> 5;
    int lr = lane & 15, kh = lane >> 4;
    const float* sq = ws + WS_SQ;
    int ibase = i0 + wave * 16 + 8 * kh;
    float sqi[8];
#pragma unroll
    for (int v = 0; v < 8; ++v) sqi[v] = sq[ibase + v];

    // bandwidth = sum(l2)/(n^2-n) / KERNEL_MUL^(KERNEL_NUM//2), KERNEL_MUL=2 -> /4
    float nn = (float)NTOT * (float)NTOT - (float)NTOT;
    float bandwidth = (ws[WS_BWSUM] / nn) * 0.25f;
    float c0 = 1.0f / (2.0f * bandwidth * bandwidth);   // coeff for q=0; scales by 1/4 per q

    float lsum = 0.f;
#pragma unroll
    for (int t = 0; t < NSUB; ++t) {
        float sqj = sq[j0 + t * 16 + lr];
#pragma unroll
        for (int v = 0; v < 8; ++v) {
            float l2 = fmaxf(sqi[v] + sqj - 2.0f * acc[t][v], 0.0f);
            float cc = c0, k = 0.f;
#pragma unroll
            for (int q = 0; q < 5; ++q) { k += __expf(-l2 * cc); cc *= 0.25f; }
            lsum += k;
        }
    }
    // quadrant sign: block lies entirely within one half on each axis (4096 % 128 == 0)
    float sgn = ((i0 < NS) == (j0 < NS)) ? 1.0f : -1.0f;
    block_reduce_store(lsum * sgn, lds, ws + WS_P2 + blockIdx.y * gridDim.x + blockIdx.x);
}

// ------------------------------------------------- deterministic reductions
__global__ __launch_bounds__(256) void reduce4096(const float* __restrict__ in,
                                                  float* __restrict__ out, float scale) {
    __shared__ float scratch[8];
    float s = 0.f;
    for (int i = threadIdx.x; i < 4096; i += 256) s += in[i];
    int lane = threadIdx.x & 31, wave = threadIdx.x >> 5;
    for (int off = 16; off; off >>= 1) s += __shfl_down(s, off, 32);
    if (lane == 0) scratch[wave] = s;
    __syncthreads();
    if (threadIdx.x == 0) {
        float t = 0.f;
#pragma unroll
        for (int w = 0; w < 8; ++w) t += scratch[w];
        out[0] = t * scale;
    }
}

extern "C" void kernel_launch(void* const* d_in, const int* in_sizes, int n_in,
                              void* d_out, int out_size, void* d_ws, size_t ws_size,
                              hipStream_t stream) {
    (void)in_sizes; (void)n_in; (void)out_size; (void)ws_size;
    const float* src = (const float*)d_in[0];
    const float* tgt = (const float*)d_in[1];
    float* ws  = (float*)d_ws;
    float* out = (float*)d_out;

    rownorm_kernel<<<NTOT / 8, 256, 0, stream>>>(src, tgt, ws);

    dim3 grid(NTOT / TILE, NTOT / TILE);
    size_t smem = (size_t)2 * TILE * LD * sizeof(float);   // ~260 KB of the 320 KB WGP LDS

    suml2_kernel<<<grid, 256, smem, stream>>>(src, tgt, ws);
    reduce4096<<<1, 256, 0, stream>>>(ws + WS_P1, ws + WS_BWSUM, 1.0f);
    mmd_kernel<<<grid, 256, smem, stream>>>(src, tgt, ws);
    reduce4096<<<1, 256, 0, stream>>>(ws + WS_P2, out, 1.0f / ((float)NS * (float)NS));
}